// RNN_76046690943147
// MI455X (gfx1250) — compile-verified
//
#include <hip/hip_runtime.h>

// Problem dims (fixed by the reference)
#define B_ 64
#define S_ 1024
#define I_ 512
#define H_ 512
#define WROW 516  // padded LDS row stride (floats): conflict-free b64 fragment loads

typedef float v2f __attribute__((ext_vector_type(2)));
typedef float v8f __attribute__((ext_vector_type(8)));

// D = A(16x4 f32) * B(4x16 f32) + C(16x16 f32), full fp32 precision.
__device__ __forceinline__ v8f wmma_f32(v2f a, v2f b, v8f c) {
  return __builtin_amdgcn_wmma_f32_16x16x4_f32(
      /*neg_a=*/false, a, /*neg_b=*/false, b,
      /*c_mod=*/(short)0, c, /*reuse_a=*/false, /*reuse_b=*/false);
}

// ---------------------------------------------------------------------------
// Phase 1: x_proj[M=B*S, H] = inputs[M, I] @ W_ih^T + (bias_ih + bias_hh)
// Block: 256 threads = 8 waves. Block covers 8 M-tiles x 2 N-tiles.
// Each wave: 1 M-tile x 2 N-tiles -> each A fragment feeds 2 WMMAs
// (halves global traffic, 2 independent accumulation chains for ILP).
// W_ih slice (32 rows x 512) staged once in LDS (66 KB, fine on CDNA5).
// ---------------------------------------------------------------------------
__global__ __launch_bounds__(256) void xproj_kernel(
    const float* __restrict__ inputs, const float* __restrict__ Wih,
    const float* __restrict__ bih, const float* __restrict__ bhh,
    float* __restrict__ out) {
  __shared__ float Wlds[32 * WROW];

  const int npair = blockIdx.x & 15;  // 16 pairs of N tiles
  const int mblk  = blockIdx.x >> 4;  // 512 blocks of 8 M-tiles
  const int tid   = threadIdx.x;
  const int nBase0 = npair * 32;      // rows nBase0..nBase0+31 staged

  // Stage W_ih[nBase0..nBase0+31][0..511] into LDS (row-major, padded rows).
  for (int idx = tid; idx < 32 * I_; idx += 256) {
    const int j = idx >> 9;      // /512
    const int k = idx & 511;
    Wlds[j * WROW + k] = Wih[(nBase0 + j) * I_ + k];
  }
  __syncthreads();

  const int wave = tid >> 5;          // 0..7 -> M tile within block
  const int lane = tid & 31;
  const int l16  = lane & 15;
  const int hi   = lane >> 4;         // 0 or 1 (half-wave)
  const int koff = hi * 2;            // A/B upper half-wave holds K+2,K+3
  const long mBase = ((long)mblk * 8 + wave) * 16;

  const float* Arow = inputs + (mBase + l16) * (long)I_;
  const float  bsum0 = bih[nBase0 + l16] + bhh[nBase0 + l16];
  const float  bsum1 = bih[nBase0 + 16 + l16] + bhh[nBase0 + 16 + l16];

  v8f c0 = {}, c1 = {};
#pragma unroll 4
  for (int kk = 0; kk < I_; kk += 4) {
    const int k0 = kk + koff;
    v2f a  = *(const v2f*)(Arow + k0);                        // A[m][k0..k0+1]
    v2f b0 = *(const v2f*)(&Wlds[l16 * WROW + k0]);           // N-tile 0
    v2f b1 = *(const v2f*)(&Wlds[(16 + l16) * WROW + k0]);    // N-tile 1
    c0 = wmma_f32(a, b0, c0);
    c1 = wmma_f32(a, b1, c1);
  }

  // C layout: VGPR v -> row (v + hi*8), col = l16 within tile.
  float* orow0 = out + (mBase + hi * 8) * (long)H_ + nBase0 + l16;
  float* orow1 = orow0 + 16;
#pragma unroll
  for (int v = 0; v < 8; ++v) {
    orow0[(long)v * H_] = c0[v] + bsum0;
    orow1[(long)v * H_] = c1[v] + bsum1;
  }
}

// ---------------------------------------------------------------------------
// Phase 2: sequential scan, persistent co-resident grid of 32 blocks.
// Block n owns N-tile n (W_hh 16x512 slice in LDS, staged once).
// 4 waves = 4 M-tiles. Per step, per wave: K=512 split into 4 independent
// accumulator chains (4-way WMMA ILP on the serial critical path); x_proj
// tile loads issued before the K loop and folded in at the end; relu;
// in-place overwrite of out[:,t,:]; device-scope barrier.
// ---------------------------------------------------------------------------
__global__ __launch_bounds__(128) void scan_kernel(
    const float* __restrict__ h0, const float* __restrict__ Whh,
    float* __restrict__ out, unsigned* __restrict__ ctr) {
  __shared__ float Wlds[16 * WROW];

  const int nBase = blockIdx.x * 16;  // 32 blocks == 32 N tiles
  const int tid   = threadIdx.x;

  for (int idx = tid; idx < 16 * H_; idx += 128) {
    const int j = idx >> 9;
    const int k = idx & 511;
    Wlds[j * WROW + k] = Whh[(nBase + j) * H_ + k];
  }
  __syncthreads();

  const int wave  = tid >> 5;   // 0..3 -> M tile (batch rows)
  const int lane  = tid & 31;
  const int l16   = lane & 15;
  const int hi    = lane >> 4;
  const int koff  = hi * 2;
  const int mBase = wave * 16;
  const int mA    = mBase + l16;              // A-fragment row = batch index
  const unsigned nblocks = gridDim.x;
  const float* Brow = &Wlds[l16 * WROW];

  for (int t = 0; t < S_; ++t) {
    // A source: h_{t-1} (out slice t-1, already relu'd) or h0 at t==0.
    const float* Arow = (t == 0)
        ? (h0 + (long)mA * H_)
        : (out + ((long)mA * S_ + (t - 1)) * (long)H_);

    // Issue x_proj tile loads first; they retire under the WMMA loop.
    float* xrow = out + (((long)(mBase + hi * 8)) * S_ + t) * (long)H_ + nBase + l16;
    float xp[8];
#pragma unroll
    for (int v = 0; v < 8; ++v) xp[v] = xrow[(long)v * S_ * H_];

    // 4 independent accumulation chains over K quarters.
    v8f c0 = {}, c1 = {}, c2 = {}, c3 = {};
#pragma unroll 2
    for (int kk = 0; kk < 128; kk += 4) {
      const int k0 = kk + koff;
      v2f a0 = *(const v2f*)(Arow + k0);
      v2f a1 = *(const v2f*)(Arow + 128 + k0);
      v2f a2 = *(const v2f*)(Arow + 256 + k0);
      v2f a3 = *(const v2f*)(Arow + 384 + k0);
      v2f b0 = *(const v2f*)(Brow + k0);
      v2f b1 = *(const v2f*)(Brow + 128 + k0);
      v2f b2 = *(const v2f*)(Brow + 256 + k0);
      v2f b3 = *(const v2f*)(Brow + 384 + k0);
      c0 = wmma_f32(a0, b0, c0);
      c1 = wmma_f32(a1, b1, c1);
      c2 = wmma_f32(a2, b2, c2);
      c3 = wmma_f32(a3, b3, c3);
    }

    // h_t = relu(xp + sum of chains); overwrite out[:,t,:] in place.
#pragma unroll
    for (int v = 0; v < 8; ++v) {
      float hv = xp[v] + ((c0[v] + c1[v]) + (c2[v] + c3[v]));
      xrow[(long)v * S_ * H_] = hv > 0.0f ? hv : 0.0f;
    }

    // Device-scope step barrier (32 co-resident blocks).
    __threadfence();
    __syncthreads();
    if (tid == 0) {
      __hip_atomic_fetch_add(ctr, 1u, __ATOMIC_RELEASE, __HIP_MEMORY_SCOPE_AGENT);
      const unsigned target = nblocks * (unsigned)(t + 1);
      while (__hip_atomic_load(ctr, __ATOMIC_ACQUIRE, __HIP_MEMORY_SCOPE_AGENT) < target) {
        __builtin_amdgcn_s_sleep(1);
      }
    }
    __syncthreads();
  }
}

__global__ void init_ctr_kernel(unsigned* c) {
  if (threadIdx.x == 0 && blockIdx.x == 0) *c = 0u;
}

__global__ void hfinal_kernel(const float* __restrict__ out, float* __restrict__ hf) {
  const int i = blockIdx.x * blockDim.x + threadIdx.x;  // over B*H
  if (i < B_ * H_) {
    const int b = i / H_;
    const int n = i - b * H_;
    hf[i] = out[((long)b * S_ + (S_ - 1)) * (long)H_ + n];
  }
}

extern "C" void kernel_launch(void* const* d_in, const int* in_sizes, int n_in,
                              void* d_out, int out_size, void* d_ws, size_t ws_size,
                              hipStream_t stream) {
  const float* inputs = (const float*)d_in[0];  // [B,S,I]
  const float* h0     = (const float*)d_in[1];  // [1,B,H]
  const float* Wih    = (const float*)d_in[2];  // [H,I]
  const float* Whh    = (const float*)d_in[3];  // [H,H]
  const float* bih    = (const float*)d_in[4];  // [H]
  const float* bhh    = (const float*)d_in[5];  // [H]
  float* out = (float*)d_out;                   // [B,S,H] outputs ++ [1,B,H] h_final
  unsigned* ctr = (unsigned*)d_ws;              // barrier counter

  // Phase 1: 4096 M-tiles / 8 per block * 16 N-tile pairs = 8192 blocks.
  xproj_kernel<<<dim3(512 * 16), dim3(256), 0, stream>>>(inputs, Wih, bih, bhh, out);

  // Phase 2: reset barrier, run persistent scan.
  init_ctr_kernel<<<1, 32, 0, stream>>>(ctr);
  scan_kernel<<<dim3(32), dim3(128), 0, stream>>>(h0, Whh, out, ctr);

  // h_final = out[:, S-1, :]
  float* hf = out + (long)B_ * S_ * H_;
  hfinal_kernel<<<dim3((B_ * H_ + 255) / 256), dim3(256), 0, stream>>>(out, hf);
}